// PhysORD_1219770712796
// MI455X (gfx1250) — compile-verified
//
#include <hip/hip_runtime.h>
#include <math.h>

typedef __attribute__((ext_vector_type(2))) float v2f;
typedef __attribute__((ext_vector_type(8))) float v8f;

#define ACT_S 68  // activation row stride (== 4 mod 64 -> conflict-free banks)
#define FW_S 68   // f-MLP weight row stride (padded from 64)

// ---------------------------------------------------------------------------
// fp32-exact WMMA: D(16x16,f32) = A(16x4,f32) * B(4x16,f32) + C
// ---------------------------------------------------------------------------
__device__ __forceinline__ v8f wmma_f32_16x16x4(v2f a, v2f b, v8f c) {
  return __builtin_amdgcn_wmma_f32_16x16x4_f32(false, a, false, b, (short)0, c,
                                               false, false);
}

// Branch-free tanh: tanh(x) = sign(x) * (1 - 2/(exp(2|x|)+1)).
// v_exp_f32 + v_rcp_f32, no EXEC divergence; ~2-3 ulp absolute error.
__device__ __forceinline__ float fast_tanh(float x) {
  const float ax = fabsf(x);
  const float e = __builtin_amdgcn_exp2f(ax * 2.8853900817779268f);  // exp(2ax)
  const float t = 1.0f - 2.0f * __builtin_amdgcn_rcpf(e + 1.0f);
  return copysignf(t, x);
}

// One dense layer for one 16-row tile held in LDS (row stride ACT_S floats).
// X: [16][ACT_S] input (cols 0..4*KC-1 valid, padded cols zero)
// W: [4*KC][SW] row-major zero-padded weights in LDS
// In-place X==Y is safe: all A-fragments are in VGPRs before any store, and
// per-wave LDS ops are in-order.
template <int KC, int NT, int SW, bool ACT>
__device__ __forceinline__ void wmma_layer(const float* Xin, float* Yout,
                                           const float* W, const float* bias,
                                           int lane) {
  const int m = lane & 15;
  const int khalf = (lane >> 4) << 1;  // lanes 16..31 hold K+2 / K+3
  const int mb = (lane >> 4) << 3;     // C/D rows: 0..7 | 8..15
  v2f afr[KC];
#pragma unroll
  for (int kc = 0; kc < KC; ++kc) {
    const int kb = kc * 4 + khalf;
    v2f a;
    a.x = Xin[m * ACT_S + kb];
    a.y = Xin[m * ACT_S + kb + 1];
    afr[kc] = a;
  }
#pragma unroll
  for (int nt = 0; nt < NT; ++nt) {
    const int n = nt * 16 + m;
    v8f acc = {};
#pragma unroll
    for (int kc = 0; kc < KC; ++kc) {
      const int kb = kc * 4 + khalf;
      v2f b;
      b.x = W[kb * SW + n];
      b.y = W[(kb + 1) * SW + n];
      acc = wmma_f32_16x16x4(afr[kc], b, acc);
    }
    const float bn = bias[n];
#pragma unroll
    for (int r = 0; r < 8; ++r) {
      float y = acc[r] + bn;
      if (ACT) y = fast_tanh(y);
      Yout[(mb + r) * ACT_S + n] = y;
    }
  }
}

__device__ __forceinline__ void dv_mlp(float* Xt, const float* W1,
                                       const float* b1, const float* W2,
                                       const float* b2, const float* W3,
                                       const float* b3, int lane) {
  wmma_layer<4, 1, 16, true>(Xt, Xt, W1, b1, lane);   // 12(->16) -> 10(->16)
  wmma_layer<4, 1, 16, true>(Xt, Xt, W2, b2, lane);   // 10(->16) -> 10(->16)
  wmma_layer<4, 1, 16, false>(Xt, Xt, W3, b3, lane);  // 10(->16) -> 12(->16)
}

// Cramer's-rule 3x3 solve: x = A^{-1} b
__device__ __forceinline__ void solve3(const float A[3][3], const float b[3],
                                       float x[3]) {
  const float c00 = A[1][1] * A[2][2] - A[1][2] * A[2][1];
  const float c01 = A[1][0] * A[2][2] - A[1][2] * A[2][0];
  const float c02 = A[1][0] * A[2][1] - A[1][1] * A[2][0];
  const float det = A[0][0] * c00 - A[0][1] * c01 + A[0][2] * c02;
  const float inv = 1.0f / det;
  const float d0 = b[0] * c00 - A[0][1] * (b[1] * A[2][2] - A[1][2] * b[2]) +
                   A[0][2] * (b[1] * A[2][1] - A[1][1] * b[2]);
  const float d1 = A[0][0] * (b[1] * A[2][2] - A[1][2] * b[2]) - b[0] * c01 +
                   A[0][2] * (A[1][0] * b[2] - b[1] * A[2][0]);
  const float d2 = A[0][0] * (A[1][1] * b[2] - b[1] * A[2][1]) -
                   A[0][1] * (A[1][0] * b[2] - b[1] * A[2][0]) + b[0] * c02;
  x[0] = d0 * inv;
  x[1] = d1 * inv;
  x[2] = d2 * inv;
}

// One PhysORD step. Each wave owns 32 bodies (one per lane); the MLPs run as
// two 16-row WMMA tiles so scalar physics uses all 32 lanes.
__global__ __launch_bounds__(128) void physord_step_kernel(
    const float* __restrict__ prev, const float* __restrict__ actsrc,
    float* __restrict__ next, const float* __restrict__ dv_w1,
    const float* __restrict__ dv_b1, const float* __restrict__ dv_w2,
    const float* __restrict__ dv_b2, const float* __restrict__ dv_w3,
    const float* __restrict__ dv_b3, const float* __restrict__ f_w1,
    const float* __restrict__ f_b1, const float* __restrict__ f_w2,
    const float* __restrict__ f_b2, const float* __restrict__ f_w3,
    const float* __restrict__ f_b3) {
  // ---- LDS: zero-padded weights + per-wave 32x68 activation tile ----
  __shared__ float sW1[16 * FW_S];  // f layer1: K 13->16, stride padded
  __shared__ float sW2[64 * FW_S];  // f layer2
  __shared__ float sW3[64 * 16];    // f layer3: N 6->16
  __shared__ float sb1[64], sb2[64], sb3[16];
  __shared__ float sdW1[256], sdW2[256], sdW3[256];  // dv layers padded 16x16
  __shared__ float sdb1[16], sdb2[16], sdb3[16];
  __shared__ float sAct[4][32 * ACT_S];

  const int tid = threadIdx.x;
  for (int i = tid; i < 16 * FW_S; i += 128) {
    int r = i / FW_S, c = i % FW_S;
    sW1[i] = (r < 13 && c < 64) ? f_w1[r * 64 + c] : 0.0f;
  }
  for (int i = tid; i < 64 * FW_S; i += 128) {
    int r = i / FW_S, c = i % FW_S;
    sW2[i] = (c < 64) ? f_w2[r * 64 + c] : 0.0f;
  }
  for (int i = tid; i < 64 * 16; i += 128) {
    int r = i >> 4, c = i & 15;
    sW3[i] = (c < 6) ? f_w3[r * 6 + c] : 0.0f;
  }
  for (int i = tid; i < 64; i += 128) {
    sb1[i] = f_b1[i];
    sb2[i] = f_b2[i];
  }
  for (int i = tid; i < 256; i += 128) {
    int r = i >> 4, c = i & 15;
    sdW1[i] = (r < 12 && c < 10) ? dv_w1[r * 10 + c] : 0.0f;
    sdW2[i] = (r < 10 && c < 10) ? dv_w2[r * 10 + c] : 0.0f;
    sdW3[i] = (r < 10 && c < 12) ? dv_w3[r * 12 + c] : 0.0f;
  }
  if (tid < 16) {
    sb3[tid] = (tid < 6) ? f_b3[tid] : 0.0f;
    sdb1[tid] = (tid < 10) ? dv_b1[tid] : 0.0f;
    sdb2[tid] = (tid < 10) ? dv_b2[tid] : 0.0f;
    sdb3[tid] = (tid < 12) ? dv_b3[tid] : 0.0f;
  }
  __syncthreads();

  const int wave = tid >> 5;
  const int lane = tid & 31;
  float* X = sAct[wave];          // [32][ACT_S]
  float* X0 = X;                  // rows 0..15  (WMMA tile 0)
  float* X1 = X + 16 * ACT_S;     // rows 16..31 (WMMA tile 1)
  float* xr = X + lane * ACT_S;   // this lane's body row
  const int row = blockIdx.x * 128 + wave * 32 + lane;
  const float Jd[3] = {400.0f, 400.0f, 1000.0f};
  const float invM = 1.0f / 900.0f;

  // Per-body state in registers (one body per lane).
  float qx[3], R[9], vk[3], om[3], sk4[4], rpm[4];
  float ef[6], fxm[3], fRm[3], pR[3], Mk[3];
  float F[3][3], Rn[9], Rfxm[3], dVx[3], qxn[3];

  // ---- load state, stage f-MLP input [v(3),omega(3),u(3),v_gap(4)] ----
  {
    const float* p = prev + (size_t)row * 29;
    qx[0] = p[0]; qx[1] = p[1]; qx[2] = p[2];
#pragma unroll
    for (int i = 0; i < 9; ++i) R[i] = p[3 + i];
    vk[0] = p[12]; vk[1] = p[13]; vk[2] = p[14];
    om[0] = p[15]; om[1] = p[16]; om[2] = p[17];
    sk4[0] = p[18]; sk4[1] = p[19]; sk4[2] = p[20]; sk4[3] = p[21];
    rpm[0] = p[22]; rpm[1] = p[23]; rpm[2] = p[24]; rpm[3] = p[25];
    const float vsum = sqrtf(vk[0] * vk[0] + vk[1] * vk[1] + vk[2] * vk[2]);
    xr[0] = vk[0]; xr[1] = vk[1]; xr[2] = vk[2];
    xr[3] = om[0]; xr[4] = om[1]; xr[5] = om[2];
    xr[6] = p[26]; xr[7] = p[27]; xr[8] = p[28];
#pragma unroll
    for (int j = 0; j < 4; ++j) xr[9 + j] = vsum - rpm[j] * (2.0f / 60.0f);
    xr[13] = 0.0f; xr[14] = 0.0f; xr[15] = 0.0f;
  }

  // ---- f-MLP (13->64->64->6), two 16-row tiles per layer ----
  wmma_layer<4, 4, FW_S, true>(X0, X0, sW1, sb1, lane);
  wmma_layer<4, 4, FW_S, true>(X1, X1, sW1, sb1, lane);
  wmma_layer<16, 4, FW_S, true>(X0, X0, sW2, sb2, lane);
  wmma_layer<16, 4, FW_S, true>(X1, X1, sW2, sb2, lane);
  wmma_layer<16, 1, 16, false>(X0, X0, sW3, sb3, lane);
  wmma_layer<16, 1, 16, false>(X1, X1, sW3, sb3, lane);

  // ---- collect ef, stage qk for dv-MLP #1 ----
#pragma unroll
  for (int i = 0; i < 6; ++i) ef[i] = xr[i];
  xr[0] = qx[0]; xr[1] = qx[1]; xr[2] = qx[2];
#pragma unroll
  for (int i = 0; i < 9; ++i) xr[3 + i] = R[i];
  xr[12] = 0.0f; xr[13] = 0.0f; xr[14] = 0.0f; xr[15] = 0.0f;
  dv_mlp(X0, sdW1, sdb1, sdW2, sdb2, sdW3, sdb3, lane);
  dv_mlp(X1, sdW1, sdb1, sdW2, sdb2, sdW3, sdb3, lane);

  // ---- physics: moment, implicit Newton solve, Cayley map, q update ----
  {
    float dVR[9];
#pragma unroll
    for (int i = 0; i < 3; ++i) dVx[i] = xr[i];
#pragma unroll
    for (int i = 0; i < 9; ++i) dVR[i] = xr[3 + i];
    auto SMf = [&](int i, int k) {
      float s = 0.0f;
#pragma unroll
      for (int j = 0; j < 3; ++j)
        s += dVR[3 * j + i] * R[3 * j + k] - R[3 * j + i] * dVR[3 * j + k];
      return s;
    };
    Mk[0] = SMf(2, 1); Mk[1] = SMf(0, 2); Mk[2] = SMf(1, 0);
    float a[3], v3[3] = {0.0f, 0.0f, 0.0f};
#pragma unroll
    for (int i = 0; i < 3; ++i) {
      pR[i] = om[i] * Jd[i];
      fRm[i] = 0.05f * ef[i];      // C*H*fR ( == (1-C)*H*fR )
      fxm[i] = 0.05f * ef[3 + i];
      a[i] = 0.1f * pR[i] + 0.005f * Mk[i] + 0.1f * fRm[i];
    }
#pragma unroll
    for (int it = 0; it < 3; ++it) {
      const float aTv = a[0] * v3[0] + a[1] * v3[1] + a[2] * v3[2];
      const float phi[3] = {
          a[0] + (a[1] * v3[2] - a[2] * v3[1]) + v3[0] * aTv - 2.0f * v3[0] * Jd[0],
          a[1] + (a[2] * v3[0] - a[0] * v3[2]) + v3[1] * aTv - 2.0f * v3[1] * Jd[1],
          a[2] + (a[0] * v3[1] - a[1] * v3[0]) + v3[2] * aTv - 2.0f * v3[2] * Jd[2]};
      const float D[3][3] = {
          {aTv - 2.0f * Jd[0] + v3[0] * a[0], -a[2] + v3[0] * a[1], a[1] + v3[0] * a[2]},
          {a[2] + v3[1] * a[0], aTv - 2.0f * Jd[1] + v3[1] * a[1], -a[0] + v3[1] * a[2]},
          {-a[1] + v3[2] * a[0], a[0] + v3[2] * a[1], aTv - 2.0f * Jd[2] + v3[2] * a[2]}};
      float dx[3];
      solve3(D, phi, dx);
      v3[0] -= dx[0]; v3[1] -= dx[1]; v3[2] -= dx[2];
    }
    const float Sv[3][3] = {{0.0f, -v3[2], v3[1]},
                            {v3[2], 0.0f, -v3[0]},
                            {-v3[1], v3[0], 0.0f}};
    const float u2 = 1.0f + v3[0] * v3[0] + v3[1] * v3[1] + v3[2] * v3[2];
    const float iu2 = 1.0f / u2;
#pragma unroll
    for (int i = 0; i < 3; ++i)
#pragma unroll
      for (int j = 0; j < 3; ++j) {
        const float ss =
            Sv[i][0] * Sv[0][j] + Sv[i][1] * Sv[1][j] + Sv[i][2] * Sv[2][j];
        F[i][j] = (((i == j) ? u2 : 0.0f) + 2.0f * Sv[i][j] + 2.0f * ss) * iu2;
      }
#pragma unroll
    for (int i = 0; i < 3; ++i) {
#pragma unroll
      for (int j = 0; j < 3; ++j)
        Rn[3 * i + j] =
            R[3 * i] * F[0][j] + R[3 * i + 1] * F[1][j] + R[3 * i + 2] * F[2][j];
      Rfxm[i] = R[3 * i] * fxm[0] + R[3 * i + 1] * fxm[1] + R[3 * i + 2] * fxm[2];
      qxn[i] = qx[i] + 0.1f * (vk[i] * 900.0f) * invM + 0.1f * Rfxm[i] * invM -
               0.005f * dVx[i] * invM;
    }
    // stage qk_next for dv-MLP #2
    xr[0] = qxn[0]; xr[1] = qxn[1]; xr[2] = qxn[2];
#pragma unroll
    for (int i = 0; i < 9; ++i) xr[3 + i] = Rn[i];
    xr[12] = 0.0f; xr[13] = 0.0f; xr[14] = 0.0f; xr[15] = 0.0f;
  }
  dv_mlp(X0, sdW1, sdb1, sdW2, sdb2, sdW3, sdb3, lane);
  dv_mlp(X1, sdW1, sdb1, sdW2, sdb2, sdW3, sdb3, lane);

  // ---- momentum updates and output ----
  {
    float dVxn[3], dVRn[9];
#pragma unroll
    for (int i = 0; i < 3; ++i) dVxn[i] = xr[i];
#pragma unroll
    for (int i = 0; i < 9; ++i) dVRn[i] = xr[3 + i];
    auto SMn = [&](int i, int k) {
      float s = 0.0f;
#pragma unroll
      for (int j = 0; j < 3; ++j)
        s += dVRn[3 * j + i] * Rn[3 * j + k] - Rn[3 * j + i] * dVRn[3 * j + k];
      return s;
    };
    float Mkn[3] = {SMn(2, 1), SMn(0, 2), SMn(1, 0)};
    float* o = next + (size_t)row * 29;
#pragma unroll
    for (int i = 0; i < 3; ++i) {
      const float t1 = F[0][i] * pR[0] + F[1][i] * pR[1] + F[2][i] * pR[2];
      const float t2 = F[0][i] * Mk[0] + F[1][i] * Mk[1] + F[2][i] * Mk[2];
      const float t3 = F[0][i] * fRm[0] + F[1][i] * fRm[1] + F[2][i] * fRm[2];
      const float pRn = t1 + 0.05f * t2 + 0.05f * Mkn[i] + t3 + 0.05f * ef[i];
      const float rnfp = Rn[3 * i] * (0.05f * ef[3]) +
                         Rn[3 * i + 1] * (0.05f * ef[4]) +
                         Rn[3 * i + 2] * (0.05f * ef[5]);
      const float pxn = -0.05f * dVx[i] - 0.05f * dVxn[i] + Rfxm[i] + rnfp;
      o[i] = qxn[i];
      o[12 + i] = pxn * invM + vk[i];
      o[15 + i] = pRn / Jd[i];
    }
#pragma unroll
    for (int i = 0; i < 9; ++i) o[3 + i] = Rn[i];
#pragma unroll
    for (int i = 0; i < 4; ++i) {
      o[18 + i] = sk4[i];
      o[22 + i] = rpm[i];
    }
    const float* asrc = actsrc + (size_t)row * 29;
    o[26] = asrc[26];
    o[27] = asrc[27];
    o[28] = asrc[28];  // scan: nx[:, -3:] = action
  }
}

__global__ void copy_step0_kernel(const float* __restrict__ src,
                                  float* __restrict__ dst, int n) {
  int i = blockIdx.x * blockDim.x + threadIdx.x;
  if (i < n) dst[i] = src[i];
}

extern "C" void kernel_launch(void* const* d_in, const int* in_sizes, int n_in,
                              void* d_out, int out_size, void* d_ws,
                              size_t ws_size, hipStream_t stream) {
  (void)n_in; (void)d_ws; (void)ws_size; (void)out_size;
  const float* traj = (const float*)d_in[1];
  const float* dv_w1 = (const float*)d_in[2];
  const float* dv_b1 = (const float*)d_in[3];
  const float* dv_w2 = (const float*)d_in[4];
  const float* dv_b2 = (const float*)d_in[5];
  const float* dv_w3 = (const float*)d_in[6];
  const float* dv_b3 = (const float*)d_in[7];
  const float* f_w1 = (const float*)d_in[8];
  const float* f_b1 = (const float*)d_in[9];
  const float* f_w2 = (const float*)d_in[10];
  const float* f_b2 = (const float*)d_in[11];
  const float* f_w3 = (const float*)d_in[12];
  const float* f_b3 = (const float*)d_in[13];

  const int STEP_NUM = 12;  // reference fixed setup
  const int B = in_sizes[1] / ((STEP_NUM + 1) * 29);
  float* out = (float*)d_out;
  const int n0 = B * 29;

  copy_step0_kernel<<<(n0 + 255) / 256, 256, 0, stream>>>(traj, out, n0);
  for (int k = 0; k < STEP_NUM; ++k) {
    physord_step_kernel<<<B / 128, 128, 0, stream>>>(
        out + (size_t)k * n0, traj + (size_t)(k + 1) * n0,
        out + (size_t)(k + 1) * n0, dv_w1, dv_b1, dv_w2, dv_b2, dv_w3, dv_b3,
        f_w1, f_b1, f_w2, f_b2, f_w3, f_b3);
  }
}